// GraphSAGE_14955076125382
// MI455X (gfx1250) — compile-verified
//
#include <hip/hip_runtime.h>
#include <hip/hip_bf16.h>

typedef float v2f __attribute__((ext_vector_type(2)));
typedef float v8f __attribute__((ext_vector_type(8)));

#if defined(__gfx1250__) && __has_builtin(__builtin_amdgcn_wmma_f32_16x16x4_f32)
#define SAGE_HAVE_WMMA 1
#endif

// D = A(16x4) * B(4x16) + C, fp32 WMMA. Wave32, EXEC must be all ones.
__device__ __forceinline__ v8f wmma_k4(v2f a, v2f b, v8f c) {
#ifdef SAGE_HAVE_WMMA
    // 8 args: (neg_a, A, neg_b, B, c_mod, C, reuse_a, reuse_b)
    return __builtin_amdgcn_wmma_f32_16x16x4_f32(
        false, a, false, b, (short)0, c, false, false);
#else
    // Shuffle-based emulation with identical per-lane layout (fallback only).
    int lane = threadIdx.x & 31;
    int half = lane >> 4;
    int n = lane & 15;
#pragma unroll
    for (int r = 0; r < 8; ++r) {
        int m = r + half * 8;
        float sum = c[r];
#pragma unroll
        for (int k = 0; k < 4; ++k) {
            int aLane = m + (k >> 1) * 16;
            int bLane = n + (k >> 1) * 16;
            float av = __shfl((k & 1) ? a.y : a.x, aLane, 32);
            float bv = __shfl((k & 1) ? b.y : b.x, bLane, 32);
            sum += av * bv;
        }
        c[r] = sum;
    }
    return c;
#endif
}

__device__ __forceinline__ float elu1(float v) {
    return v > 0.0f ? v : expm1f(v);
}

__global__ void zero_kernel(float* __restrict__ p, long long n) {
    long long i = (long long)blockIdx.x * blockDim.x + threadIdx.x;
    long long stride = (long long)gridDim.x * blockDim.x;
    for (; i < n; i += stride) p[i] = 0.0f;
}

__global__ void degree_kernel(const int* __restrict__ ei, float* __restrict__ deg, long long E) {
    long long e = (long long)blockIdx.x * blockDim.x + threadIdx.x;
    if (e < E) atomicAdd(&deg[ei[E + e]], 1.0f);
}

__global__ void invdeg_kernel(float* __restrict__ deg, int n) {
    int i = blockIdx.x * blockDim.x + threadIdx.x;
    if (i < n) deg[i] = 1.0f / fmaxf(deg[i], 1.0f);
}

// One thread per (edge, feature). Edge id is wave-uniform for D==32 (half-wave
// uniform for D==16); feature lanes give coalesced gathers + coalesced,
// L2-resident global_atomic_add_f32.
template <int D>
__global__ void scatter_kernel(const float* __restrict__ x, const int* __restrict__ ei,
                               float* __restrict__ msg, long long E) {
    long long tid = (long long)blockIdx.x * blockDim.x + threadIdx.x;
    if (tid >= E * (long long)D) return;
    int d = (int)(tid & (D - 1));
    long long e = tid / D;
    int s = ei[e];
    int t = ei[E + e];
    atomicAdd(&msg[(long long)t * D + d], x[(long long)s * D + d]);
}

// Fused SAGE transform: out = elu( (msg*invdeg) @ Wl + bias + x @ Wr ), Dout = 16.
// Weight (B) fragments + bias hoisted into registers; each wave grid-strides
// over 16-node tiles so the steady-state loop is A-loads + WMMA only.
template <int DIN>
__global__ void sage_wmma_kernel(const float* __restrict__ x, const float* __restrict__ msg,
                                 const float* __restrict__ invdeg,
                                 const float* __restrict__ Wl, const float* __restrict__ Wr,
                                 const float* __restrict__ bias, float* __restrict__ out,
                                 int ntiles, int nnodes) {
    constexpr int NK = DIN / 4;
    int wave = blockIdx.x * (blockDim.x >> 5) + (threadIdx.x >> 5);
    int nwaves = gridDim.x * (blockDim.x >> 5);
    int lane = threadIdx.x & 31;
    int half = lane >> 4;
    int mn = lane & 15;

    // Hoist B fragments (per-lane column mn, K-pair (half*2, half*2+1) per chunk).
    v2f bL[NK], bR[NK];
#pragma unroll
    for (int c = 0; c < NK; ++c) {
        int kb = c * 4 + half * 2;
        bL[c].x = Wl[kb * 16 + mn];
        bL[c].y = Wl[(kb + 1) * 16 + mn];
        bR[c].x = Wr[kb * 16 + mn];
        bR[c].y = Wr[(kb + 1) * 16 + mn];
    }
    float bv = bias[mn];

    for (int tile = wave; tile < ntiles; tile += nwaves) {   // wave-uniform loop
        int rowA = tile * 16 + mn;
        if (rowA >= nnodes) rowA = nnodes - 1;               // safety (N%16==0 here)
        float sc = invdeg[rowA];
        const float* xr = x + (long long)rowA * DIN;
        const float* mr = msg + (long long)rowA * DIN;
        v8f acc = {0.f, 0.f, 0.f, 0.f, 0.f, 0.f, 0.f, 0.f};
#pragma unroll
        for (int c = 0; c < NK; ++c) {
            int kb = c * 4 + half * 2;
            v2f aA; aA.x = mr[kb] * sc; aA.y = mr[kb + 1] * sc;
            acc = wmma_k4(aA, bL[c], acc);
            v2f aX; aX.x = xr[kb]; aX.y = xr[kb + 1];
            acc = wmma_k4(aX, bR[c], acc);
        }
#pragma unroll
        for (int r = 0; r < 8; ++r) {
            int row = tile * 16 + half * 8 + r;
            float v = elu1(acc[r] + bv);
            if (row < nnodes) out[(long long)row * 16 + mn] = v;
        }
    }
}

// out = act(in @ W + bias); blockIdx.y selects a 16-wide column tile.
// B fragments + bias register-resident; grid-stride over node tiles.
template <int DIN, int DOUT, bool DO_ELU>
__global__ void linear_wmma_kernel(const float* __restrict__ in, const float* __restrict__ W,
                                   const float* __restrict__ bias, float* __restrict__ out,
                                   int ntiles, int nnodes) {
    constexpr int NK = DIN / 4;
    int wave = blockIdx.x * (blockDim.x >> 5) + (threadIdx.x >> 5);
    int nwaves = gridDim.x * (blockDim.x >> 5);
    int ct = blockIdx.y;
    int lane = threadIdx.x & 31;
    int half = lane >> 4;
    int mn = lane & 15;
    int col = ct * 16 + mn;

    v2f bF[NK];
#pragma unroll
    for (int c = 0; c < NK; ++c) {
        int kb = c * 4 + half * 2;
        bF[c].x = W[kb * DOUT + col];
        bF[c].y = W[(kb + 1) * DOUT + col];
    }
    float bv = bias[col];

    for (int tile = wave; tile < ntiles; tile += nwaves) {   // wave-uniform loop
        int rowA = tile * 16 + mn;
        if (rowA >= nnodes) rowA = nnodes - 1;
        const float* ir = in + (long long)rowA * DIN;
        v8f acc = {0.f, 0.f, 0.f, 0.f, 0.f, 0.f, 0.f, 0.f};
#pragma unroll
        for (int c = 0; c < NK; ++c) {
            int kb = c * 4 + half * 2;
            v2f a; a.x = ir[kb]; a.y = ir[kb + 1];
            acc = wmma_k4(a, bF[c], acc);
        }
#pragma unroll
        for (int r = 0; r < 8; ++r) {
            int row = tile * 16 + half * 8 + r;
            float v = acc[r] + bv;
            if (DO_ELU) v = elu1(v);
            if (row < nnodes) out[(long long)row * DOUT + col] = v;
        }
    }
}

// Final 64 -> 4 projection: trivial VALU kernel (memory-bound, no 16-col tile).
__global__ void final_kernel(const float* __restrict__ in, const float* __restrict__ W,
                             const float* __restrict__ b, float* __restrict__ out, int nnodes) {
    int t = blockIdx.x * blockDim.x + threadIdx.x;
    if (t >= nnodes * 4) return;
    int n = t >> 2;
    int c = t & 3;
    const float* ir = in + (long long)n * 64;
    float s = b[c];
#pragma unroll
    for (int k = 0; k < 64; ++k) s += ir[k] * W[k * 4 + c];
    out[t] = s;
}

extern "C" void kernel_launch(void* const* d_in, const int* in_sizes, int n_in,
                              void* d_out, int out_size, void* d_ws, size_t ws_size,
                              hipStream_t stream) {
    (void)n_in; (void)out_size; (void)ws_size;
    const float* x   = (const float*)d_in[0];
    const int*   ei  = (const int*)d_in[1];
    const float* Wl0 = (const float*)d_in[2];
    const float* Wr0 = (const float*)d_in[3];
    const float* bl0 = (const float*)d_in[4];
    const float* Wl1 = (const float*)d_in[5];
    const float* Wr1 = (const float*)d_in[6];
    const float* bl1 = (const float*)d_in[7];
    const float* Wl2 = (const float*)d_in[8];
    const float* Wr2 = (const float*)d_in[9];
    const float* bl2 = (const float*)d_in[10];
    const float* LW0 = (const float*)d_in[11];
    const float* LB0 = (const float*)d_in[12];
    const float* LW1 = (const float*)d_in[13];
    const float* LB1 = (const float*)d_in[14];
    const float* LW2 = (const float*)d_in[15];
    const float* LB2 = (const float*)d_in[16];
    const float* LW3 = (const float*)d_in[17];
    const float* LB3 = (const float*)d_in[18];

    const int N = in_sizes[0] / 32;
    const long long E = (long long)in_sizes[1] / 2;

    float* ws  = (float*)d_ws;
    float* deg = ws;                          // N
    float* msg = deg + N;                     // N*32
    float* hA  = msg + (size_t)N * 32;        // N*16
    float* hB  = hA + (size_t)N * 16;         // N*16
    float* lA  = hB + (size_t)N * 16;         // N*64
    float* lB  = lA + (size_t)N * 64;         // N*64

    const int ntiles = (N + 15) / 16;
    const dim3 blk(256);                      // 8 waves per block (wave32)
    // ~3072 waves: full occupancy with ~2 tiles/wave of register weight reuse.
    const int gemmBlocks = 384;
    const int edgeBlocks = (int)((E + 255) / 256);

    // Degrees (shared by all conv layers), then reciprocal.
    zero_kernel<<<1024, blk, 0, stream>>>(deg, N);
    degree_kernel<<<edgeBlocks, blk, 0, stream>>>(ei, deg, E);
    invdeg_kernel<<<(N + 255) / 256, blk, 0, stream>>>(deg, N);

    // Conv 0: 32 -> 16
    zero_kernel<<<4096, blk, 0, stream>>>(msg, (long long)N * 32);
    scatter_kernel<32><<<(int)((E * 32 + 255) / 256), blk, 0, stream>>>(x, ei, msg, E);
    sage_wmma_kernel<32><<<gemmBlocks, blk, 0, stream>>>(x, msg, deg, Wl0, Wr0, bl0, hA, ntiles, N);

    // Conv 1: 16 -> 16
    zero_kernel<<<4096, blk, 0, stream>>>(msg, (long long)N * 16);
    scatter_kernel<16><<<(int)((E * 16 + 255) / 256), blk, 0, stream>>>(hA, ei, msg, E);
    sage_wmma_kernel<16><<<gemmBlocks, blk, 0, stream>>>(hA, msg, deg, Wl1, Wr1, bl1, hB, ntiles, N);

    // Conv 2: 16 -> 16
    zero_kernel<<<4096, blk, 0, stream>>>(msg, (long long)N * 16);
    scatter_kernel<16><<<(int)((E * 16 + 255) / 256), blk, 0, stream>>>(hB, ei, msg, E);
    sage_wmma_kernel<16><<<gemmBlocks, blk, 0, stream>>>(hB, msg, deg, Wl2, Wr2, bl2, hA, ntiles, N);

    // MLP head
    linear_wmma_kernel<16, 64, true><<<dim3(gemmBlocks, 4), blk, 0, stream>>>(hA, LW0, LB0, lA, ntiles, N);
    linear_wmma_kernel<64, 64, true><<<dim3(gemmBlocks, 4), blk, 0, stream>>>(lA, LW1, LB1, lB, ntiles, N);
    linear_wmma_kernel<64, 64, true><<<dim3(gemmBlocks, 4), blk, 0, stream>>>(lB, LW2, LB2, lA, ntiles, N);
    final_kernel<<<(N * 4 + 255) / 256, blk, 0, stream>>>(lA, LW3, LB3, (float*)d_out, N);
}